// MetadataTapas_3573412790675
// MI455X (gfx1250) — compile-verified
//
#include <hip/hip_runtime.h>
#include <math.h>

// Shapes fixed by the reference setup
#define BB 32
#define SS 1024
#define HH 768
#define FF 128
#define NSEG 129          // F+1
#define PP 256
#define NCAT 32           // 24 live head columns padded to 2 N-tiles of 16
#define HCH 64            // H-chunk per segsum block

typedef __attribute__((ext_vector_type(2))) float v2f;
typedef __attribute__((ext_vector_type(8))) float v8f;

// ---------------- segment counts (per batch, LDS float atomics) -------------
__global__ void count_kernel(const int* __restrict__ col_ids, float* __restrict__ cnts) {
    __shared__ float c[NSEG];
    int b = blockIdx.x;
    for (int i = threadIdx.x; i < NSEG; i += blockDim.x) c[i] = 0.0f;
    __syncthreads();
    for (int s = threadIdx.x; s < SS; s += blockDim.x)
        atomicAdd(&c[col_ids[b * SS + s]], 1.0f);
    __syncthreads();
    for (int i = threadIdx.x; i < NSEG; i += blockDim.x)
        cnts[b * NSEG + i] = c[i];
}

// ---------------- segment sums: block = (h-chunk, batch), LDS accum ---------
__global__ void segsum_kernel(const float* __restrict__ emb,
                              const int* __restrict__ col_ids,
                              float* __restrict__ sums) {
    __shared__ float acc[NSEG * HCH];   // 33 KB
    int b  = blockIdx.y;
    int h0 = blockIdx.x * HCH;
    for (int i = threadIdx.x; i < NSEG * HCH; i += blockDim.x) acc[i] = 0.0f;
    __syncthreads();
    int d = threadIdx.x & (HCH - 1);
    int t = threadIdx.x / HCH;          // 4 tokens in flight
    for (int s = t; s < SS; s += blockDim.x / HCH) {
        int col = col_ids[b * SS + s];
        float v = emb[((size_t)(b * SS + s)) * HH + h0 + d];
        atomicAdd(&acc[col * HCH + d], v);   // ds_add_f32
    }
    __syncthreads();
    for (int i = threadIdx.x; i < NSEG * HCH; i += blockDim.x) {
        int seg = i / HCH, dd = i % HCH;
        sums[((size_t)b * NSEG + seg) * HH + h0 + dd] = acc[i];
    }
}

// ---------------- in-place mean: sums[b][1+f][:] /= max(cnt,1) --------------
__global__ void fieldnorm_kernel(float* __restrict__ sums, const float* __restrict__ cnts) {
    size_t idx = (size_t)blockIdx.x * blockDim.x + threadIdx.x;
    if (idx >= (size_t)BB * FF * HH) return;
    int    h  = (int)(idx % HH);
    size_t rf = idx / HH;
    int    f  = (int)(rf % FF);
    int    b  = (int)(rf / FF);
    size_t row = (size_t)b * NSEG + 1 + f;
    float  c   = cnts[b * NSEG + 1 + f];
    sums[row * HH + h] /= fmaxf(c, 1.0f);
}

// ---------------- pack 6 head weights/biases into 768x32 (+32) --------------
__global__ void pack_kernel(const float* Wm, const float* bm, const float* Wa, const float* ba,
                            const float* Wd, const float* bd, const float* Wms, const float* bms,
                            const float* Wk, const float* bk, const float* Wt, const float* bt,
                            float* __restrict__ Wcat, float* __restrict__ bcat) {
    int idx = blockIdx.x * blockDim.x + threadIdx.x;
    if (idx < NCAT) {
        int n = idx; float v = 0.0f;
        if      (n < 2)  v = bm[n];
        else if (n < 11) v = ba[n - 2];
        else if (n < 13) v = bd[n - 11];
        else if (n < 15) v = bms[n - 13];
        else if (n < 17) v = bk[n - 15];
        else if (n < 24) v = bt[n - 17];
        bcat[n] = v;
    }
    if (idx >= HH * NCAT) return;
    int k = idx >> 5, n = idx & 31;
    float v = 0.0f;
    if      (n < 2)  v = Wm[k * 2 + n];
    else if (n < 11) v = Wa[k * 9 + (n - 2)];
    else if (n < 13) v = Wd[k * 2 + (n - 11)];
    else if (n < 15) v = Wms[k * 2 + (n - 13)];
    else if (n < 17) v = Wk[k * 2 + (n - 15)];
    else if (n < 24) v = Wt[k * 7 + (n - 17)];
    Wcat[idx] = v;
}

// ---------------- WMMA GEMM: (4096 x 768) x (768 x 32), f32 16x16x4 --------
// 1 wave per 16-row M-tile; 2 accumulators cover N=0..31.
__global__ void gemm_kernel(const float* __restrict__ sums,
                            const float* __restrict__ Wcat,
                            float* __restrict__ gout) {
    int lane  = threadIdx.x & 31;
    int wave  = threadIdx.x >> 5;
    int mtile = blockIdx.x * 4 + wave;          // 256 tiles total
    int m0    = mtile * 16;
    int khalf = lane >> 4;                      // 0: K=k0,k0+1  1: K=k0+2,k0+3
    int ml    = lane & 15;                      // A: M index / B: N index

    // field row for this lane's A fragment (tiles never cross batch boundary)
    int b = (m0 + ml) >> 7, f = (m0 + ml) & 127;
    const float* arow = sums + ((size_t)b * NSEG + 1 + f) * HH + 2 * khalf;

    v8f acc0 = {}, acc1 = {};
    for (int k0 = 0; k0 < HH; k0 += 4) {
        const float2 a2 = *(const float2*)(arow + k0);
        v2f a = { a2.x, a2.y };
        const float* brow = Wcat + (k0 + 2 * khalf) * NCAT;
        v2f b0 = { brow[ml],          brow[NCAT + ml] };
        v2f b1 = { brow[16 + ml],     brow[NCAT + 16 + ml] };
        acc0 = __builtin_amdgcn_wmma_f32_16x16x4_f32(false, a, false, b0, (short)0, acc0, false, false);
        acc1 = __builtin_amdgcn_wmma_f32_16x16x4_f32(false, a, false, b1, (short)0, acc1, false, false);
    }
    // C/D layout: VGPR j -> lanes0-15 M=j, lanes16-31 M=8+j; N=lane&15
    for (int j = 0; j < 8; ++j) {
        int M = j + 8 * khalf;
        gout[(size_t)(m0 + M) * NCAT + ml]      = acc0[j];
        gout[(size_t)(m0 + M) * NCAT + 16 + ml] = acc1[j];
    }
}

// ---------------- bias + log_softmax + scatter to d_out ---------------------
__global__ void heads_kernel(const float* __restrict__ gemm,
                             const float* __restrict__ bcat,
                             float* __restrict__ out) {
    int idx = blockIdx.x * blockDim.x + threadIdx.x;
    if (idx >= BB * FF * 6) return;
    int row = idx / 6, head = idx % 6;
    int base = 0, c = 2; size_t ooff = 0;
    switch (head) {
        case 0: base = 0;  c = 2; ooff = 0;     break;  // msr_res        (out 1)
        case 1: base = 2;  c = 9; ooff = 8192;  break;  // agg_score_res  (out 2)
        case 2: base = 11; c = 2; ooff = 53248; break;  // dim_score_res  (out 4)
        case 3: base = 13; c = 2; ooff = 45056; break;  // msr_score_res  (out 3)
        case 4: base = 15; c = 2; ooff = 61440; break;  // key_score_res  (out 5)
        case 5: base = 17; c = 7; ooff = 86016; break;  // msr_type_res   (out 7)
    }
    float l[9]; float m = -1e30f;
    for (int i = 0; i < c; ++i) {
        l[i] = gemm[(size_t)row * NCAT + base + i] + bcat[base + i];
        m = fmaxf(m, l[i]);
    }
    float s = 0.0f;
    for (int i = 0; i < c; ++i) s += expf(l[i] - m);
    float lse = m + logf(s);
    for (int i = 0; i < c; ++i) out[ooff + (size_t)row * c + i] = l[i] - lse;
}

// ---------------- pair head: wave per (b,p), 1536-dot + wave32 reduce -------
__global__ void pair_kernel(const float* __restrict__ sums,
                            const int* __restrict__ pidx,
                            const float* __restrict__ Wp,
                            const float* __restrict__ bp,
                            float* __restrict__ out) {
    int wid  = (int)((blockIdx.x * (size_t)blockDim.x + threadIdx.x) >> 5);
    int lane = threadIdx.x & 31;
    if (wid >= BB * PP) return;
    int b = wid >> 8, p = wid & 255;
    int i1 = pidx[(b * PP + p) * 2 + 0];
    int i2 = pidx[(b * PP + p) * 2 + 1];
    const float* f1 = sums + ((size_t)b * NSEG + 1 + i1) * HH;
    const float* f2 = sums + ((size_t)b * NSEG + 1 + i2) * HH;
    float s0 = 0.0f, s1 = 0.0f;
    for (int k = lane; k < 2 * HH; k += 32) {
        float x = (k < HH) ? f1[k] : f2[k - HH];
        s0 += x * Wp[k * 2 + 0];
        s1 += x * Wp[k * 2 + 1];
    }
    for (int off = 16; off > 0; off >>= 1) {
        s0 += __shfl_xor(s0, off, 32);
        s1 += __shfl_xor(s1, off, 32);
    }
    if (lane == 0) {
        float l0 = s0 + bp[0], l1 = s1 + bp[1];
        float m  = fmaxf(l0, l1);
        float lse = m + logf(expf(l0 - m) + expf(l1 - m));
        out[69632 + (size_t)(b * PP + p) * 2 + 0] = l0 - lse;
        out[69632 + (size_t)(b * PP + p) * 2 + 1] = l1 - lse;
    }
}

extern "C" void kernel_launch(void* const* d_in, const int* in_sizes, int n_in,
                              void* d_out, int out_size, void* d_ws, size_t ws_size,
                              hipStream_t stream) {
    const float* emb     = (const float*)d_in[0];
    const int*   col_ids = (const int*)d_in[1];
    const int*   pidx    = (const int*)d_in[2];
    // d_in[3] = n_fields (hardcoded 128)
    const float* Wm  = (const float*)d_in[4],  *bm  = (const float*)d_in[5];
    const float* Wa  = (const float*)d_in[6],  *ba  = (const float*)d_in[7];
    const float* Wd  = (const float*)d_in[8],  *bd  = (const float*)d_in[9];
    const float* Wms = (const float*)d_in[10], *bms = (const float*)d_in[11];
    const float* Wk  = (const float*)d_in[12], *bk  = (const float*)d_in[13];
    const float* Wp  = (const float*)d_in[14], *bp  = (const float*)d_in[15];
    const float* Wt  = (const float*)d_in[16], *bt  = (const float*)d_in[17];
    float* out = (float*)d_out;

    auto align256 = [](size_t x) { return (x + 255) & ~(size_t)255; };
    char* ws = (char*)d_ws;
    size_t off = 0;
    float* sums = (float*)(ws + off); off = align256(off + (size_t)BB * NSEG * HH * 4);
    float* cnts = (float*)(ws + off); off = align256(off + (size_t)BB * NSEG * 4);
    float* Wcat = (float*)(ws + off); off = align256(off + (size_t)HH * NCAT * 4);
    float* bcat = (float*)(ws + off); off = align256(off + (size_t)NCAT * 4);
    float* gout = (float*)(ws + off); off = align256(off + (size_t)BB * FF * NCAT * 4);
    (void)ws_size; (void)n_in; (void)in_sizes; (void)out_size;

    // 1) segment counts and sums (LDS accumulation, pure stores out)
    count_kernel<<<BB, 256, 0, stream>>>(col_ids, cnts);
    segsum_kernel<<<dim3(HH / HCH, BB), 256, 0, stream>>>(emb, col_ids, sums);

    // 2) in-place mean -> field rows live at sums[b*129 + 1 + f]
    {
        size_t n = (size_t)BB * FF * HH;
        fieldnorm_kernel<<<(unsigned)((n + 255) / 256), 256, 0, stream>>>(sums, cnts);
    }

    // 3) pack head weights, one fused WMMA GEMM, then softmax-finalize
    pack_kernel<<<(HH * NCAT + 255) / 256, 256, 0, stream>>>(
        Wm, bm, Wa, ba, Wd, bd, Wms, bms, Wk, bk, Wt, bt, Wcat, bcat);
    gemm_kernel<<<(BB * FF / 16) / 4, 128, 0, stream>>>(sums, Wcat, gout);
    heads_kernel<<<(BB * FF * 6 + 255) / 256, 256, 0, stream>>>(gout, bcat, out);

    // 4) pair head (wave per pair)
    pair_kernel<<<(BB * PP * 32 + 255) / 256, 256, 0, stream>>>(sums, pidx, Wp, bp, out);
}